// MatchingLayer_62620623176370
// MI455X (gfx1250) — compile-verified
//
#include <hip/hip_runtime.h>
#include <math.h>

#define EPS 1e-8f
#define MIN_VAL -1e7f

typedef __attribute__((ext_vector_type(2))) float v2f;
typedef __attribute__((ext_vector_type(8))) float v8f;

// B=32, L=128, H=100, P=20, out channels per side = 105
// Channel map: 0=cos_max 1=cos_mean 2=f1 3..22=fP 23..42=max 43..62=mean
//              63=am1 64..83=amP 84=ax1 85..104=axP

// ---------------------------------------------------------------- wsq
__global__ void wsq_kernel(const float* w0, const float* w1, const float* w2,
                           const float* w3, float* wsq) {
  int idx = blockIdx.x * 256 + threadIdx.x;
  if (idx >= 8000) return;
  int q = idx / 2000, r = idx % 2000;
  const float* w = (q == 0) ? w0 : (q == 1) ? w1 : (q == 2) ? w2 : w3;
  float v = w[r];
  wsq[idx] = v * v;
}

// ---------------------------------------------------------------- prep
// masked contexts, row norms, mask counts, last-token rows
__global__ void prep_kernel(const float* ctxp, const int* maskp,
                            const float* ctxh, const int* maskh,
                            float* cp, float* ch, float* npc, float* nhc,
                            float* cnt, float* cp_last, float* ch_last) {
  int b = blockIdx.x, i = threadIdx.x;
  __shared__ int smp[128], smh[128];
  __shared__ int sidx[2];
  int mp = maskp[b * 128 + i], mh = maskh[b * 128 + i];
  smp[i] = mp; smh[i] = mh;
  float fp = mp ? 1.f : 0.f, fh = mh ? 1.f : 0.f;
  const float* prow = ctxp + (size_t)(b * 128 + i) * 100;
  const float* hrow = ctxh + (size_t)(b * 128 + i) * 100;
  float* cprow = cp + (size_t)(b * 128 + i) * 100;
  float* chrow = ch + (size_t)(b * 128 + i) * 100;
  float s2p = 0.f, s2h = 0.f;
  for (int k = 0; k < 100; ++k) {
    float vp = prow[k] * fp, vh = hrow[k] * fh;
    cprow[k] = vp; chrow[k] = vh;
    s2p += vp * vp; s2h += vh * vh;
  }
  npc[b * 128 + i] = sqrtf(s2p);
  nhc[b * 128 + i] = sqrtf(s2h);
  __syncthreads();
  if (i == 0) {
    int sp = 0, sh = 0;
    for (int t = 0; t < 128; ++t) { sp += smp[t]; sh += smh[t]; }
    cnt[b * 4 + 0] = (float)sp; cnt[b * 4 + 1] = (float)sh;
    cnt[b * 4 + 2] = sp > 0 ? 1.f : 0.f; cnt[b * 4 + 3] = sh > 0 ? 1.f : 0.f;
    sidx[0] = sp > 0 ? sp - 1 : 0; sidx[1] = sh > 0 ? sh - 1 : 0;
  }
  __syncthreads();
  if (i < 100) {
    cp_last[b * 100 + i] = cp[(size_t)(b * 128 + sidx[0]) * 100 + i];
    ch_last[b * 100 + i] = ch[(size_t)(b * 128 + sidx[1]) * 100 + i];
  }
}

// ---------------------------------------------------------------- weighted norms (w1)
__global__ void wnorm_kernel(const float* cp, const float* ch, const float* wsq1,
                             float* n1w, float* n2w) {
  int b = blockIdx.x, p = blockIdx.y, i = threadIdx.x;
  const float* w = wsq1 + p * 100;
  const float* a = cp + (size_t)(b * 128 + i) * 100;
  const float* c = ch + (size_t)(b * 128 + i) * 100;
  float s1 = 0.f, s2 = 0.f;
  for (int k = 0; k < 100; ++k) {
    float wk = w[k];
    s1 += wk * a[k] * a[k];
    s2 += wk * c[k] * c[k];
  }
  n1w[(b * 20 + p) * 128 + i] = sqrtf(s1);
  n2w[(b * 20 + p) * 128 + i] = sqrtf(s2);
}

// ---------------------------------------------------------------- cos GEMM (WMMA) + pooled reductions
// block = 256 (8 waves), grid.x = B. Wave w owns i-tile w; loops 8 j-tiles.
__global__ void cos_wmma_kernel(const float* cp, const float* ch,
                                const float* npc, const float* nhc,
                                const int* maskp, const int* maskh, const float* cnt,
                                float* cosm, float* rsum, float* csum,
                                float* outp, float* outh) {
  int b = blockIdx.x;
  int tid = threadIdx.x, w = tid >> 5, lane = tid & 31, half = lane >> 4, lq = lane & 15;
  __shared__ float sCMax[8][128], sCSum[8][128];
  int i0 = w * 16;
  float npr[8]; int mpr[8];
  for (int v = 0; v < 8; ++v) {
    int r = i0 + v + half * 8;
    npr[v] = npc[b * 128 + r];
    mpr[v] = maskp[b * 128 + r] > 0;
  }
  float rmax[8], rsm[8];
  for (int v = 0; v < 8; ++v) { rmax[v] = MIN_VAL; rsm[v] = 0.f; }
  const float* cpb = cp + (size_t)b * 128 * 100;
  const float* chb = ch + (size_t)b * 128 * 100;
  const float* arow = cpb + (i0 + lq) * 100 + (half ? 2 : 0);
  for (int jt = 0; jt < 8; ++jt) {
    int j0 = jt * 16;
    const float* brow = chb + (j0 + lq) * 100 + (half ? 2 : 0);
    v8f acc = {0.f, 0.f, 0.f, 0.f, 0.f, 0.f, 0.f, 0.f};
    for (int ks = 0; ks < 25; ++ks) {
      v2f a, bb;
      a.x = arow[ks * 4]; a.y = arow[ks * 4 + 1];
      bb.x = brow[ks * 4]; bb.y = brow[ks * 4 + 1];
      acc = __builtin_amdgcn_wmma_f32_16x16x4_f32(false, a, false, bb,
                                                  (short)0, acc, false, false);
    }
    int jc = j0 + lq;
    float nh = nhc[b * 128 + jc];
    int mh = maskh[b * 128 + jc] > 0;
    float cmaxp = MIN_VAL, csump = 0.f;
    for (int v = 0; v < 8; ++v) {
      int r = i0 + v + half * 8;
      float c = acc[v] / fmaxf(npr[v] * nh, EPS);
      cosm[(size_t)(b * 128 + r) * 128 + jc] = c;
      rmax[v] = fmaxf(rmax[v], mh ? c : MIN_VAL);
      rsm[v] += c;  // c == 0 exactly where mh == 0
      cmaxp = fmaxf(cmaxp, mpr[v] ? c : MIN_VAL);
      csump += c;
    }
    cmaxp = fmaxf(cmaxp, __shfl_xor(cmaxp, 16, 32));
    csump += __shfl_xor(csump, 16, 32);
    if (lane < 16) { sCMax[w][jc] = cmaxp; sCSum[w][jc] = csump; }
  }
  float sump = cnt[b * 4 + 0], sumh = cnt[b * 4 + 1];
  float anyp = cnt[b * 4 + 2], anyh = cnt[b * 4 + 3];
  for (int v = 0; v < 8; ++v) {
    float rm = rmax[v], rs = rsm[v];
    for (int m = 1; m < 16; m <<= 1) {
      rm = fmaxf(rm, __shfl_xor(rm, m, 32));
      rs += __shfl_xor(rs, m, 32);
    }
    rmax[v] = rm; rsm[v] = rs;
  }
  if (lq == 0) {
    for (int v = 0; v < 8; ++v) {
      int r = i0 + v + half * 8;
      float mp = mpr[v] ? 1.f : 0.f;
      float* o = outp + (size_t)(b * 128 + r) * 105;
      o[0] = rmax[v] * (mp * anyh);
      o[1] = (mp * rsm[v]) / fmaxf(mp * sumh, EPS);
      rsum[b * 128 + r] = rsm[v];
    }
  }
  __syncthreads();
  if (tid < 128) {
    float cm = MIN_VAL, cs = 0.f;
    for (int q = 0; q < 8; ++q) { cm = fmaxf(cm, sCMax[q][tid]); cs += sCSum[q][tid]; }
    float mh = maskh[b * 128 + tid] > 0 ? 1.f : 0.f;
    float* o = outh + (size_t)(b * 128 + tid) * 105;
    o[0] = cm * (mh * anyp);
    o[1] = (mh * cs) / fmaxf(mh * sump, EPS);
    csum[b * 128 + tid] = cs;
  }
}

// ---------------------------------------------------------------- maxpool match GEMMs (WMMA), one block per (b,p)
__global__ void maxpool_wmma_kernel(const float* cp, const float* ch, const float* wsq1,
                                    const float* n1w, const float* n2w,
                                    const int* maskp, const int* maskh, const float* cnt,
                                    float* outp, float* outh) {
  int b = blockIdx.x, p = blockIdx.y;
  int tid = threadIdx.x, w = tid >> 5, lane = tid & 31, half = lane >> 4, lq = lane & 15;
  __shared__ float sW[100];
  __shared__ float sCMax[8][128], sCSum[8][128];
  for (int t = tid; t < 100; t += 256) sW[t] = wsq1[p * 100 + t];
  __syncthreads();
  int i0 = w * 16;
  const float* n1b = n1w + (b * 20 + p) * 128;
  const float* n2b = n2w + (b * 20 + p) * 128;
  float npr[8]; int mpr[8];
  for (int v = 0; v < 8; ++v) {
    int r = i0 + v + half * 8;
    npr[v] = n1b[r];
    mpr[v] = maskp[b * 128 + r] > 0;
  }
  float rmax[8], rsm[8];
  for (int v = 0; v < 8; ++v) { rmax[v] = MIN_VAL; rsm[v] = 0.f; }
  int ka = half ? 2 : 0;
  const float* arow = cp + (size_t)(b * 128 + i0 + lq) * 100 + ka;
  for (int jt = 0; jt < 8; ++jt) {
    int j0 = jt * 16;
    const float* brow = ch + (size_t)(b * 128 + j0 + lq) * 100 + ka;
    v8f acc = {0.f, 0.f, 0.f, 0.f, 0.f, 0.f, 0.f, 0.f};
    for (int ks = 0; ks < 25; ++ks) {
      int kk = ks * 4 + ka;
      v2f a, bb;
      a.x = arow[ks * 4] * sW[kk];
      a.y = arow[ks * 4 + 1] * sW[kk + 1];
      bb.x = brow[ks * 4]; bb.y = brow[ks * 4 + 1];
      acc = __builtin_amdgcn_wmma_f32_16x16x4_f32(false, a, false, bb,
                                                  (short)0, acc, false, false);
    }
    int jc = j0 + lq;
    float nh = n2b[jc];
    int mh = maskh[b * 128 + jc] > 0;
    float cmaxp = MIN_VAL, csump = 0.f;
    for (int v = 0; v < 8; ++v) {
      float c = acc[v] / fmaxf(npr[v] * nh, EPS);
      rmax[v] = fmaxf(rmax[v], mh ? c : MIN_VAL);
      rsm[v] += c;
      cmaxp = fmaxf(cmaxp, mpr[v] ? c : MIN_VAL);
      csump += c;
    }
    cmaxp = fmaxf(cmaxp, __shfl_xor(cmaxp, 16, 32));
    csump += __shfl_xor(csump, 16, 32);
    if (lane < 16) { sCMax[w][jc] = cmaxp; sCSum[w][jc] = csump; }
  }
  float sump = cnt[b * 4 + 0], sumh = cnt[b * 4 + 1];
  float anyp = cnt[b * 4 + 2], anyh = cnt[b * 4 + 3];
  for (int v = 0; v < 8; ++v) {
    float rm = rmax[v], rs = rsm[v];
    for (int m = 1; m < 16; m <<= 1) {
      rm = fmaxf(rm, __shfl_xor(rm, m, 32));
      rs += __shfl_xor(rs, m, 32);
    }
    rmax[v] = rm; rsm[v] = rs;
  }
  if (lq == 0) {
    for (int v = 0; v < 8; ++v) {
      int r = i0 + v + half * 8;
      float mp = mpr[v] ? 1.f : 0.f;
      float* o = outp + (size_t)(b * 128 + r) * 105;
      o[23 + p] = rmax[v] * (mp * anyh);
      o[43 + p] = (mp * rsm[v]) / fmaxf(mp * sumh, EPS);
    }
  }
  __syncthreads();
  if (tid < 128) {
    float cm = MIN_VAL, cs = 0.f;
    for (int q = 0; q < 8; ++q) { cm = fmaxf(cm, sCMax[q][tid]); cs += sCSum[q][tid]; }
    float mh = maskh[b * 128 + tid] > 0 ? 1.f : 0.f;
    float* o = outh + (size_t)(b * 128 + tid) * 105;
    o[23 + p] = cm * (mh * anyp);
    o[43 + p] = (mh * cs) / fmaxf(mh * sump, EPS);
  }
}

// ---------------------------------------------------------------- attentive-mean GEMMs (WMMA)
// amh[b,i,k] = (sum_j cos[i,j]*ch[j,k]) / max(rsum[i],EPS)
// amp[b,j,k] = (sum_i cos[i,j]*cp[i,k]) / max(csum[j],EPS)
__global__ void attmean_wmma_kernel(const float* cp, const float* ch, const float* cosm,
                                    const float* rsum, const float* csum,
                                    float* amh, float* amp) {
  int b = blockIdx.x;
  int tid = threadIdx.x, w = tid >> 5, lane = tid & 31, half = lane >> 4, lq = lane & 15;
  int r0 = w * 16;
  const float* cosb = cosm + (size_t)b * 128 * 128;
  // phase 1: amh (rows = i)
  for (int nt = 0; nt < 7; ++nt) {
    int kc = nt * 16 + lq;
    bool kok = kc < 100;
    v8f acc = {0.f, 0.f, 0.f, 0.f, 0.f, 0.f, 0.f, 0.f};
    for (int js = 0; js < 32; ++js) {
      int ja = js * 4 + (half ? 2 : 0);
      v2f a, bb;
      a.x = cosb[(r0 + lq) * 128 + ja];
      a.y = cosb[(r0 + lq) * 128 + ja + 1];
      bb.x = kok ? ch[(size_t)(b * 128 + ja) * 100 + kc] : 0.f;
      bb.y = kok ? ch[(size_t)(b * 128 + ja + 1) * 100 + kc] : 0.f;
      acc = __builtin_amdgcn_wmma_f32_16x16x4_f32(false, a, false, bb,
                                                  (short)0, acc, false, false);
    }
    for (int v = 0; v < 8; ++v) {
      int r = r0 + v + half * 8;
      if (kok) amh[(size_t)(b * 128 + r) * 100 + kc] = acc[v] / fmaxf(rsum[b * 128 + r], EPS);
    }
  }
  // phase 2: amp (rows = j); A[m=j][kk=i] = cos[i][j]
  for (int nt = 0; nt < 7; ++nt) {
    int kc = nt * 16 + lq;
    bool kok = kc < 100;
    v8f acc = {0.f, 0.f, 0.f, 0.f, 0.f, 0.f, 0.f, 0.f};
    for (int is = 0; is < 32; ++is) {
      int ia = is * 4 + (half ? 2 : 0);
      v2f a, bb;
      a.x = cosb[ia * 128 + (r0 + lq)];
      a.y = cosb[(ia + 1) * 128 + (r0 + lq)];
      bb.x = kok ? cp[(size_t)(b * 128 + ia) * 100 + kc] : 0.f;
      bb.y = kok ? cp[(size_t)(b * 128 + ia + 1) * 100 + kc] : 0.f;
      acc = __builtin_amdgcn_wmma_f32_16x16x4_f32(false, a, false, bb,
                                                  (short)0, acc, false, false);
    }
    for (int v = 0; v < 8; ++v) {
      int r = r0 + v + half * 8;
      if (kok) amp[(size_t)(b * 128 + r) * 100 + kc] = acc[v] / fmaxf(csum[b * 128 + r], EPS);
    }
  }
}

// ---------------------------------------------------------------- max-attentive (max-times, VALU)
__global__ void attmax_kernel(const float* cp, const float* ch, const float* cosm,
                              const int* maskp, const int* maskh, const float* cnt,
                              float* axh, float* axp) {
  int b = blockIdx.x, row = blockIdx.y, side = blockIdx.z, k = threadIdx.x;
  if (k >= 100) return;
  if (side == 0) {
    float mp = maskp[b * 128 + row] > 0 ? 1.f : 0.f;
    float anyh = cnt[b * 4 + 3];
    float vmax = MIN_VAL;
    const float* crow = cosm + (size_t)(b * 128 + row) * 128;
    for (int j = 0; j < 128; ++j) {
      if (maskh[b * 128 + j] > 0) {
        float v = ch[(size_t)(b * 128 + j) * 100 + k] * crow[j];
        vmax = fmaxf(vmax, v);
      }
    }
    axh[(size_t)(b * 128 + row) * 100 + k] = vmax * (mp * anyh);
  } else {
    float mh = maskh[b * 128 + row] > 0 ? 1.f : 0.f;
    float anyp = cnt[b * 4 + 2];
    float vmax = MIN_VAL;
    for (int i = 0; i < 128; ++i) {
      if (maskp[b * 128 + i] > 0) {
        float v = cp[(size_t)(b * 128 + i) * 100 + k] * cosm[(size_t)(b * 128 + i) * 128 + row];
        vmax = fmaxf(vmax, v);
      }
    }
    axp[(size_t)(b * 128 + row) * 100 + k] = vmax * (mh * anyp);
  }
}

// ---------------------------------------------------------------- per-(b,i) weighted cosine matches
__global__ void match_kernel(const float* cp, const float* ch,
                             const float* cp_last, const float* ch_last,
                             const float* amh, const float* amp,
                             const float* axh, const float* axp,
                             const float* wsq0, const float* wsq2, const float* wsq3,
                             float* outp, float* outh) {
  int i = blockIdx.x, b = blockIdx.y, side = blockIdx.z;
  __shared__ float sa[100], sf[100], sm[100], sx[100];
  const float *a, *bf, *bm, *bx;
  if (side == 0) {
    a = cp + (size_t)(b * 128 + i) * 100; bf = ch_last + b * 100;
    bm = amh + (size_t)(b * 128 + i) * 100; bx = axh + (size_t)(b * 128 + i) * 100;
  } else {
    a = ch + (size_t)(b * 128 + i) * 100; bf = cp_last + b * 100;
    bm = amp + (size_t)(b * 128 + i) * 100; bx = axp + (size_t)(b * 128 + i) * 100;
  }
  for (int t = threadIdx.x; t < 100; t += 64) { sa[t] = a[t]; sf[t] = bf[t]; sm[t] = bm[t]; sx[t] = bx[t]; }
  __syncthreads();
  int t = threadIdx.x;
  if (t < 63) {
    int m = t / 21, q = t % 21;
    const float* bb = (m == 0) ? sf : (m == 1) ? sm : sx;
    const float* wq = (m == 0) ? wsq0 : (m == 1) ? wsq2 : wsq3;
    float dot = 0.f, na = 0.f, nb = 0.f;
    if (q == 0) {
      for (int k = 0; k < 100; ++k) {
        float av = sa[k], bv = bb[k];
        dot += av * bv; na += av * av; nb += bv * bv;
      }
    } else {
      const float* wr = wq + (q - 1) * 100;
      for (int k = 0; k < 100; ++k) {
        float ww = wr[k], av = sa[k], bv = bb[k];
        dot += ww * av * bv; na += ww * av * av; nb += ww * bv * bv;
      }
    }
    float sim = dot / (fmaxf(sqrtf(na), EPS) * fmaxf(sqrtf(nb), EPS));
    int c = (m == 0) ? ((q == 0) ? 2 : 2 + q)
          : (m == 1) ? ((q == 0) ? 63 : 63 + q)
                     : ((q == 0) ? 84 : 84 + q);
    float* o = ((side == 0) ? outp : outh) + (size_t)(b * 128 + i) * 105;
    o[c] = sim;
  }
}

// ---------------------------------------------------------------- launch
extern "C" void kernel_launch(void* const* d_in, const int* in_sizes, int n_in,
                              void* d_out, int out_size, void* d_ws, size_t ws_size,
                              hipStream_t stream) {
  const float* ctxp = (const float*)d_in[0];
  const int*   mskp = (const int*)d_in[1];
  const float* ctxh = (const float*)d_in[2];
  const int*   mskh = (const int*)d_in[3];
  const float* w0 = (const float*)d_in[4];
  const float* w1 = (const float*)d_in[5];
  const float* w2 = (const float*)d_in[6];
  const float* w3 = (const float*)d_in[7];

  float* out  = (float*)d_out;
  float* outp = out;
  float* outh = out + (size_t)32 * 128 * 105;

  float* ws = (float*)d_ws;
  size_t off = 0;
  auto alloc = [&](size_t n) { float* p = ws + off; off += n; return p; };
  float* cp      = alloc(32 * 128 * 100);
  float* ch      = alloc(32 * 128 * 100);
  float* npc     = alloc(32 * 128);
  float* nhc     = alloc(32 * 128);
  float* cnt     = alloc(32 * 4);
  float* cp_last = alloc(32 * 100);
  float* ch_last = alloc(32 * 100);
  float* wsq     = alloc(4 * 20 * 100);
  float* n1w     = alloc(32 * 20 * 128);
  float* n2w     = alloc(32 * 20 * 128);
  float* cosm    = alloc((size_t)32 * 128 * 128);
  float* rsum    = alloc(32 * 128);
  float* csum    = alloc(32 * 128);
  float* amh     = alloc(32 * 128 * 100);
  float* amp     = alloc(32 * 128 * 100);
  float* axh     = alloc(32 * 128 * 100);
  float* axp     = alloc(32 * 128 * 100);
  float* wsq0 = wsq, *wsq1 = wsq + 2000, *wsq2 = wsq + 4000, *wsq3 = wsq + 6000;

  wsq_kernel<<<32, 256, 0, stream>>>(w0, w1, w2, w3, wsq);
  prep_kernel<<<32, 128, 0, stream>>>(ctxp, mskp, ctxh, mskh, cp, ch, npc, nhc,
                                      cnt, cp_last, ch_last);
  wnorm_kernel<<<dim3(32, 20), 128, 0, stream>>>(cp, ch, wsq1, n1w, n2w);
  cos_wmma_kernel<<<32, 256, 0, stream>>>(cp, ch, npc, nhc, mskp, mskh, cnt,
                                          cosm, rsum, csum, outp, outh);
  maxpool_wmma_kernel<<<dim3(32, 20), 256, 0, stream>>>(cp, ch, wsq1, n1w, n2w,
                                                        mskp, mskh, cnt, outp, outh);
  attmean_wmma_kernel<<<32, 256, 0, stream>>>(cp, ch, cosm, rsum, csum, amh, amp);
  attmax_kernel<<<dim3(32, 128, 2), 128, 0, stream>>>(cp, ch, cosm, mskp, mskh, cnt,
                                                      axh, axp);
  match_kernel<<<dim3(128, 32, 2), 64, 0, stream>>>(cp, ch, cp_last, ch_last,
                                                    amh, amp, axh, axp,
                                                    wsq0, wsq2, wsq3, outp, outh);
}